// Model_pixel_27212912787678
// MI455X (gfx1250) — compile-verified
//
#include <hip/hip_runtime.h>
#include <math.h>

// ---------------------------------------------------------------------------
// MPI renderer for MI455X (gfx1250, wave32).
//
// Roofline: ~37MB unique bytes (fits in 192MB L2), ~1-2 GFLOP -> gather/sort
// bound. Strategy:
//   * The K=3 einsums (ray-plane denominators and in-plane projections) are
//     computed on the matrix unit: V_WMMA_F32_16X16X4_F32 tiles of
//     G_d[pixel,plane] = sum_c dir[pixel][c]*basis[plane][c][d], 12 WMMAs per
//     wave (2 pixel-tiles x 2 plane-tiles x 3 dims), scattered to LDS.
//   * Per-pixel geometry then reads G from LDS: den = G2,
//     local_d = A0[p][d] + depth*G_d with A0 precomputed per block.
//   * Compaction: only planes with uv in [-1,1]^2 contribute (alpha=0 is a
//     compositing no-op), so we sort/sample only the ~2-3 covering planes.
//   * Plane params staged into LDS via the CDNA5 async-to-LDS path.
// ---------------------------------------------------------------------------

#define NPLANES 32
#define RES     256
#define IMG_H   480
#define IMG_W   640
#define NPIX    (IMG_H * IMG_W)
#define BLOCK   128
#define GSTRIDE (BLOCK + 4)   // pixel stride (words) in g arrays; mult. of 4

// LDS param layout (floats):
//   [0..287]   plane_basis   : p*9 + r*3 + c   (input is [P][3][3] row-major)
//   [288..383] plane_position: 288 + p*3 + c
//   [384..447] plane_wh      : 384 + p*2 + i
#define PARAMS_F 448

typedef __attribute__((ext_vector_type(2))) float v2f;
typedef __attribute__((ext_vector_type(8))) float v8f;

__device__ __forceinline__ unsigned lds_addr_of(const void* generic_ptr) {
    // gfx1250: low 32 bits of a generic pointer into __shared__ are the LDS
    // byte address (aperture bits live in [63:32]).
    return (unsigned)(size_t)generic_ptr;
}

__launch_bounds__(BLOCK)
__global__ void mpi_render_kernel(const float* __restrict__ content,   // [P][4][256][256]
                                  const float* __restrict__ basis,     // [P][3][3]
                                  const float* __restrict__ pos,       // [P][3]
                                  const float* __restrict__ wh,        // [P][2]
                                  const float* __restrict__ camR,      // [3][3]
                                  const float* __restrict__ camT,      // [3]
                                  float* __restrict__ out)             // [H][W][4]
{
    __shared__ float sp[PARAMS_F];
    __shared__ float sA0[NPLANES * 3];        // A0[p][d] = (cam_pos-pos[p]).basis[p][:,d]
    __shared__ float sInvHw[NPLANES * 2];     // 1/(wh*0.5)
    __shared__ __align__(16) float g0[NPLANES * GSTRIDE];  // G_d[plane][pixel_local]
    __shared__ __align__(16) float g1[NPLANES * GSTRIDE];
    __shared__ __align__(16) float g2[NPLANES * GSTRIDE];
    __shared__ float s_dep[NPLANES * BLOCK];
    __shared__ int   s_idx[NPLANES * BLOCK];

    const int t = threadIdx.x;

    // ---- async-stage plane params into LDS (CDNA5 async-to-LDS) ----------
    for (int i = t; i < 288; i += BLOCK) {
        unsigned lds  = lds_addr_of(&sp[i]);
        unsigned voff = (unsigned)(i * 4);
        asm volatile("global_load_async_to_lds_b32 %0, %1, %2"
                     :: "v"(lds), "v"(voff), "s"(basis) : "memory");
    }
    if (t < 96) {
        unsigned lds  = lds_addr_of(&sp[288 + t]);
        unsigned voff = (unsigned)(t * 4);
        asm volatile("global_load_async_to_lds_b32 %0, %1, %2"
                     :: "v"(lds), "v"(voff), "s"(pos) : "memory");
    }
    if (t < 64) {
        unsigned lds  = lds_addr_of(&sp[384 + t]);
        unsigned voff = (unsigned)(t * 4);
        asm volatile("global_load_async_to_lds_b32 %0, %1, %2"
                     :: "v"(lds), "v"(voff), "s"(wh) : "memory");
    }
    asm volatile("s_wait_asynccnt 0x0" ::: "memory");
    __syncthreads();

    // ---- per-block precompute: A0[p][d] and 1/(wh/2) -----------------------
    if (t < NPLANES * 3) {
        const int p = t / 3, d = t % 3;
        const float T0 = camT[0], T1 = camT[1], T2 = camT[2];
        const float cp0 = -(camR[0] * T0 + camR[3] * T1 + camR[6] * T2);
        const float cp1 = -(camR[1] * T0 + camR[4] * T1 + camR[7] * T2);
        const float cp2 = -(camR[2] * T0 + camR[5] * T1 + camR[8] * T2);
        const float* B = &sp[p * 9];
        const float q0 = cp0 - sp[288 + p * 3 + 0];
        const float q1 = cp1 - sp[288 + p * 3 + 1];
        const float q2 = cp2 - sp[288 + p * 3 + 2];
        sA0[t] = q0 * B[d] + q1 * B[3 + d] + q2 * B[6 + d];
    }
    if (t < NPLANES * 2) {
        sInvHw[t] = 1.0f / (sp[384 + t] * 0.5f);
    }
    __syncthreads();

    // ---- WMMA: G_d[pixel, plane] = sum_c dir[pixel][c] * basis[plane][c][d]
    // V_WMMA_F32_16X16X4_F32, M=16 pixels, N=16 planes, K=4 (c=3 padded 0).
    // A layout (16x4 f32): lanes 0-15 hold M=lane, v2f = (K0,K1);
    //                      lanes 16-31 hold M=lane-16, v2f = (K2, 0).
    // B layout (4x16 f32): symmetric -> lane holds column N=lane&15,
    //                      halves hold K{0,1} / K{2, 0}.
    // D layout (16x16 f32): lane holds N=lane&15, VGPR v -> M = v + 8*(lane/16).
    {
        const int lane     = t & 31;
        const int half     = lane >> 4;          // 0: K=0,1 ; 1: K=2,3
        const int mlane    = lane & 15;
        const int waveBase = t & ~31;            // waveId*32 (pixel_local base)
        const int blockPix = blockIdx.x * BLOCK;

        const float R0 = camR[0], R1 = camR[1], R2 = camR[2];
        const float R3 = camR[3], R4 = camR[4], R5 = camR[5];
        const float R6 = camR[6], R7 = camR[7], R8 = camR[8];

#pragma unroll
        for (int mt = 0; mt < 2; ++mt) {
            // A operand: dirs of pixel (waveBase + mt*16 + mlane)
            const int pix = blockPix + waveBase + mt * 16 + mlane;
            const int ppx = pix % IMG_W;
            const int ppy = pix / IMG_W;
            const float gx = ((float)ppx + 0.5f) / (float)IMG_W * 2.0f - 1.0f;
            const float gy = ((float)ppy + 0.5f) / (float)IMG_H * 2.0f - 1.0f;
            const float dw0 = gx * R0 + gy * R3 + R6;
            const float dw1 = gx * R1 + gy * R4 + R7;
            const float dw2 = gx * R2 + gy * R5 + R8;
            v2f a;
            a.x = half ? dw2 : dw0;
            a.y = half ? 0.0f : dw1;

#pragma unroll
            for (int pt = 0; pt < 2; ++pt) {
                const int plane = pt * 16 + mlane;
                const float* B = &sp[plane * 9];
#pragma unroll
                for (int d = 0; d < 3; ++d) {
                    v2f b;
                    b.x = B[half * 6 + d];              // basis[plane][half?2:0][d]
                    b.y = half ? 0.0f : B[3 + d];       // basis[plane][1][d]
                    v8f c = {};
                    v8f dres = __builtin_amdgcn_wmma_f32_16x16x4_f32(
                        /*neg_a=*/false, a, /*neg_b=*/false, b,
                        /*c_mod=*/(short)0, c, /*reuse_a=*/false, /*reuse_b=*/false);

                    // scatter: 8 consecutive pixel_locals for this plane
                    float* gd = (d == 0) ? g0 : (d == 1) ? g1 : g2;
                    const int pixbase = waveBase + mt * 16 + 8 * half;
                    float4 lo, hi;
                    lo.x = dres[0]; lo.y = dres[1]; lo.z = dres[2]; lo.w = dres[3];
                    hi.x = dres[4]; hi.y = dres[5]; hi.z = dres[6]; hi.w = dres[7];
                    *(float4*)&gd[plane * GSTRIDE + pixbase]     = lo;
                    *(float4*)&gd[plane * GSTRIDE + pixbase + 4] = hi;
                }
            }
        }
    }
    __syncthreads();

    const int gid = blockIdx.x * BLOCK + t;
    if (gid >= NPIX) return;

    // ---- pass 1: intersect all planes via G from LDS, append covering hits
    int k = 0;
    for (int p = 0; p < NPLANES; ++p) {
        const float num = -sA0[p * 3 + 2];
        float den = g2[p * GSTRIDE + t];
        if (fabsf(den) < 1e-6f) den = 1e-6f;
        const float dep = num / den;
        const float u = (sA0[p * 3 + 0] + dep * g0[p * GSTRIDE + t]) * sInvHw[p * 2 + 0];
        const float v = (sA0[p * 3 + 1] + dep * g1[p * GSTRIDE + t]) * sInvHw[p * 2 + 1];
        if (fabsf(u) <= 1.0f && fabsf(v) <= 1.0f) {
            s_dep[k * BLOCK + t] = dep;
            s_idx[k * BLOCK + t] = p;
            ++k;
        }
    }

    // ---- stable insertion sort of the k covering hits by depth ------------
    for (int i = 1; i < k; ++i) {
        const float d  = s_dep[i * BLOCK + t];
        const int   id = s_idx[i * BLOCK + t];
        int j = i - 1;
        while (j >= 0 && s_dep[j * BLOCK + t] > d) {
            s_dep[(j + 1) * BLOCK + t] = s_dep[j * BLOCK + t];
            s_idx[(j + 1) * BLOCK + t] = s_idx[j * BLOCK + t];
            --j;
        }
        s_dep[(j + 1) * BLOCK + t] = d;
        s_idx[(j + 1) * BLOCK + t] = id;
    }

    // ---- pass 2: front-to-back composite (bilinear + sigmoid per hit) -----
    float accR = 0.0f, accG = 0.0f, accB = 0.0f, accD = 0.0f, trans = 1.0f;
    for (int i = 0; i < k; ++i) {
        const float dep = s_dep[i * BLOCK + t];
        const int   p   = s_idx[i * BLOCK + t];

        const float u = (sA0[p * 3 + 0] + dep * g0[p * GSTRIDE + t]) * sInvHw[p * 2 + 0];
        const float v = (sA0[p * 3 + 1] + dep * g1[p * GSTRIDE + t]) * sInvHw[p * 2 + 1];

        // bilinear with zero-padding (matches reference gather semantics)
        const float x = (u + 1.0f) * (0.5f * (float)RES) - 0.5f;
        const float y = (v + 1.0f) * (0.5f * (float)RES) - 0.5f;
        const float xf = floorf(x), yf = floorf(y);
        const float wx = x - xf, wy = y - yf;
        const int x0 = (int)xf, y0 = (int)yf;

        const float* img = content + (size_t)p * 4 * RES * RES;
        float c0 = 0.0f, c1 = 0.0f, c2 = 0.0f, c3 = 0.0f;
#pragma unroll
        for (int dy = 0; dy < 2; ++dy) {
#pragma unroll
            for (int dx = 0; dx < 2; ++dx) {
                const int xi = x0 + dx;
                const int yi = y0 + dy;
                if (xi >= 0 && xi < RES && yi >= 0 && yi < RES) {
                    const float w = (dx ? wx : (1.0f - wx)) * (dy ? wy : (1.0f - wy));
                    const float* base = img + (size_t)yi * RES + xi;
                    c0 += w * base[0 * RES * RES];
                    c1 += w * base[1 * RES * RES];
                    c2 += w * base[2 * RES * RES];
                    c3 += w * base[3 * RES * RES];
                }
            }
        }

        const float alpha = 1.0f / (1.0f + expf(-c0));
        const float sr    = 1.0f / (1.0f + expf(-c1));
        const float sg    = 1.0f / (1.0f + expf(-c2));
        const float sb    = 1.0f / (1.0f + expf(-c3));

        const float wgt = alpha * trans;
        accD += dep * wgt;
        accR += sr * wgt;
        accG += sg * wgt;
        accB += sb * wgt;
        trans *= (1.0f - alpha);
    }

    float* o = out + (size_t)gid * 4;
    o[0] = accR;
    o[1] = accG;
    o[2] = accB;
    o[3] = accD;
}

extern "C" void kernel_launch(void* const* d_in, const int* in_sizes, int n_in,
                              void* d_out, int out_size, void* d_ws, size_t ws_size,
                              hipStream_t stream) {
    const float* content = (const float*)d_in[0];  // plane_content [32][4][256][256]
    const float* basis   = (const float*)d_in[1];  // plane_basis   [32][3][3]
    const float* pos     = (const float*)d_in[2];  // plane_position[32][3]
    const float* whp     = (const float*)d_in[3];  // plane_wh      [32][2]
    const float* camR    = (const float*)d_in[4];  // [3][3]
    const float* camT    = (const float*)d_in[5];  // [3]
    float* out = (float*)d_out;                    // [480][640][4]

    const int grid = (NPIX + BLOCK - 1) / BLOCK;   // 2400 blocks of 4 wave32s
    hipLaunchKernelGGL(mpi_render_kernel, dim3(grid), dim3(BLOCK), 0, stream,
                       content, basis, pos, whp, camR, camT, out);
}